// PixAccIoUMetric_31215822307376
// MI455X (gfx1250) — compile-verified
//
#include <hip/hip_runtime.h>

#define CLS     21
#define BATCH   16
#define PLANE   (512 * 512)                 /* floats per class plane        */
#define IMG     (CLS * PLANE)               /* floats per image              */
#define TILE_PX 128                         /* pixels per TDM tile           */
#define TPP     (PLANE / TILE_PX)           /* tiles per plane = 2048        */
#define NTILES  (BATCH * TPP)               /* 32768 tiles total             */
#define NBLOCK  2048
#define NTHREAD TILE_PX                     /* 128 threads = 4 waves         */
#define NCTR    63                          /* 3 histograms x 21 classes     */

typedef unsigned int v4u __attribute__((ext_vector_type(4)));
typedef int          v4i __attribute__((ext_vector_type(4)));
typedef int          v8i __attribute__((ext_vector_type(8)));
typedef float        v2f __attribute__((ext_vector_type(2)));
typedef float        v8f __attribute__((ext_vector_type(8)));

/* ROCm 7.2 clang-22: 5-arg builtin.  amdgpu-toolchain clang-23 (therock
 * headers, which are the only ones shipping amd_gfx1250_TDM.h): 6-arg. */
#if __has_include(<hip/amd_detail/amd_gfx1250_TDM.h>)
#define TDM_LOAD(g0, g1, z2, z3) \
    __builtin_amdgcn_tensor_load_to_lds((g0), (g1), (z2), (z3), (v8i){0,0,0,0,0,0,0,0}, 0)
#else
#define TDM_LOAD(g0, g1, z2, z3) \
    __builtin_amdgcn_tensor_load_to_lds((g0), (g1), (z2), (z3), 0)
#endif

__device__ __forceinline__ unsigned lds_off(const void* p) {
    /* generic pointer to LDS = {shared aperture, 32-bit LDS offset} */
    return (unsigned)(unsigned long long)(size_t)p;
}

/* Issue one TDM 2D tile load: [CLS rows x TILE_PX pixels], row stride PLANE.
 * D# bit packing per CDNA5 ISA ch.8 (groups 0/1; groups 2/3 zero => 2D). */
__device__ __forceinline__ void tdm_tile_load(const float* gsrc, unsigned lds_byte)
{
    const unsigned long long ga = (unsigned long long)(size_t)gsrc;

    v4u g0;
    g0.x = 1u;                                               /* count=1, user D#   */
    g0.y = lds_byte;                                         /* lds_addr           */
    g0.z = (unsigned)ga;                                     /* global_addr[31:0]  */
    g0.w = (unsigned)((ga >> 32) & 0x01FFFFFFull)            /* global_addr[56:32] */
         | (2u << 30);                                       /* type = 2 (image)   */

    const unsigned long long td0 = PLANE;                    /* tensor_dim0        */
    const unsigned long long td1 = CLS;                      /* tensor_dim1        */
    const unsigned long long s0  = PLANE;                    /* dim0 stride (elts) */
    const unsigned long long s1  = IMG;                      /* dim1 stride, benign*/

    const unsigned long long q0 = (2ull << 16)               /* data_size = 4B     */
                                | ((td0 & 0xFFFFull) << 48); /* tensor_dim0[15:0]  */
    const unsigned long long q1 = ((td0 >> 16) & 0xFFFFull)  /* tensor_dim0[31:16] */
                                | (td1 << 16)                /* tensor_dim1        */
                                | ((unsigned long long)TILE_PX << 48); /* tile_dim0 */
    const unsigned long long q2 = (unsigned long long)CLS    /* tile_dim1          */
                                | ((s0 & 0xFFFFFFFFull) << 32); /* stride0[31:0]   */
    const unsigned long long q3 = ((s0 >> 32) & 0xFFFFull)   /* stride0[47:32]     */
                                | (s1 << 16);                /* stride1[47:0]      */

    v8i g1;
    g1[0] = (int)(unsigned)q0;  g1[1] = (int)(unsigned)(q0 >> 32);
    g1[2] = (int)(unsigned)q1;  g1[3] = (int)(unsigned)(q1 >> 32);
    g1[4] = (int)(unsigned)q2;  g1[5] = (int)(unsigned)(q2 >> 32);
    g1[6] = (int)(unsigned)q3;  g1[7] = (int)(unsigned)(q3 >> 32);

    const v4i z = (v4i){0, 0, 0, 0};
    TDM_LOAD(g0, g1, z, z);
}

__global__ void iou_zero_ws(unsigned int* __restrict__ ws) {
    if (threadIdx.x < 64) ws[threadIdx.x] = 0u;
}

__global__ __launch_bounds__(NTHREAD) void iou_conf_hist_tdm(
    const float* __restrict__ outp,
    const float* __restrict__ tgtp,
    unsigned int* __restrict__ ws)
{
    __shared__ float        tiles[2][2][CLS * TILE_PX];  /* [stage][tensor] 43KB */
    __shared__ unsigned int h[64];

    const int  t      = threadIdx.x;
    const bool issuer = (t < 32);                        /* wave 0 drives TDM */
    if (t < 64) h[t] = 0u;

    /* prologue: issue stage 0 (tile = blockIdx.x) */
    {
        const unsigned g = blockIdx.x;
        const size_t   e = (size_t)(g / TPP) * IMG + (size_t)(g % TPP) * TILE_PX;
        if (issuer) {
            tdm_tile_load(outp + e, lds_off(&tiles[0][0][0]));
            tdm_tile_load(tgtp + e, lds_off(&tiles[0][1][0]));
        }
    }

    int stage = 0;
    for (unsigned g = blockIdx.x; g < NTILES; g += gridDim.x, stage ^= 1) {
        /* overlap: issue next tile into the other buffer */
        const unsigned gn   = g + gridDim.x;
        const bool     more = (gn < NTILES);
        if (more && issuer) {
            const size_t e = (size_t)(gn / TPP) * IMG + (size_t)(gn % TPP) * TILE_PX;
            tdm_tile_load(outp + e, lds_off(&tiles[stage ^ 1][0][0]));
            tdm_tile_load(tgtp + e, lds_off(&tiles[stage ^ 1][1][0]));
        }
        /* wave0: allow next tile's 2 loads outstanding, require current done */
        if (more) __builtin_amdgcn_s_wait_tensorcnt(2);
        else      __builtin_amdgcn_s_wait_tensorcnt(0);
        __syncthreads();

        const float* lo = &tiles[stage][0][0];
        const float* lt = &tiles[stage][1][0];

        int   ol = 0, tl = 0;
        bool  f  = false;
        float m  = -3.402823466e38f;
#pragma unroll
        for (int c = 0; c < CLS; ++c) {                  /* bank-conflict-free */
            const float ov = lo[(c << 7) | t];
            const float tv = lt[(c << 7) | t];
            if (!f && ov > 0.5f) { ol = c; f = true; }   /* first class > thr  */
            if (tv > m)          { m = tv; tl = c; }     /* first max (strict) */
        }
        atomicAdd(&h[21 + ol], 1u);
        atomicAdd(&h[42 + tl], 1u);
        if (ol == tl) atomicAdd(&h[tl], 1u);

        __syncthreads();   /* buffer[stage] free before it is re-issued */
    }

    if (t < NCTR) {
        const unsigned v = h[t];
        if (v) atomicAdd(&ws[t], v);
    }
}

/* Single-wave finalize: the three 21-way class reductions are dot products
 * with a ones vector, mapped onto V_WMMA_F32_16X16X4_F32.
 * A(16x4) layout (ISA 7.12.2): lanes 0-15 hold K=0 (VGPR0) / K=1 (VGPR1);
 * lane l carries class l at K=0 and class 16+l at K=1 (l<5); lanes 16-31
 * (K=2,3) carry zeros.  B = ones => D[m][n] = rowsum_m for every n, so
 * total = sum_r D[r] on lane 0 (rows 0-7) + lane 16 (rows 8-15). */
__global__ __launch_bounds__(32) void iou_finalize_wmma(
    const unsigned int* __restrict__ ws,
    float* __restrict__ res)
{
    const int l = threadIdx.x;                   /* one full wave, EXEC all-1s */

    float s0 = 0.0f, s1 = 0.0f;                  /* per-class IoU scores       */
    float t0 = 0.0f, t1 = 0.0f;                  /* per-class tp               */
    float d0 = 0.0f, d1 = 0.0f;                  /* per-class tp+fp+fn         */
    if (l < 16) {
        const unsigned tp = ws[l];
        const unsigned dn = ws[21 + l] + ws[42 + l] - tp;   /* tp+fp+fn */
        t0 = (float)tp;  d0 = (float)dn;
        s0 = (dn > 0u) ? ((float)tp / (float)dn) : 0.0f;
        if (l < CLS - 16) {
            const unsigned c   = 16 + l;
            const unsigned tp2 = ws[c];
            const unsigned dn2 = ws[21 + c] + ws[42 + c] - tp2;
            t1 = (float)tp2;  d1 = (float)dn2;
            s1 = (dn2 > 0u) ? ((float)tp2 / (float)dn2) : 0.0f;
        }
    }

    const v2f B  = (v2f){1.0f, 1.0f};            /* ones matrix, any layout    */
    const v8f Z  = (v8f){0, 0, 0, 0, 0, 0, 0, 0};
    const v2f As = (v2f){s0, s1};
    const v2f At = (v2f){t0, t1};
    const v2f Ad = (v2f){d0, d1};

    const v8f Ds = __builtin_amdgcn_wmma_f32_16x16x4_f32(
        false, As, false, B, (short)0, Z, false, false);
    const v8f Dt = __builtin_amdgcn_wmma_f32_16x16x4_f32(
        false, At, false, B, (short)0, Z, false, false);
    const v8f Dd = __builtin_amdgcn_wmma_f32_16x16x4_f32(
        false, Ad, false, B, (short)0, Z, false, false);

    float ps = 0.0f, pt = 0.0f, pd = 0.0f;       /* 8-row partials per lane    */
#pragma unroll
    for (int r = 0; r < 8; ++r) { ps += Ds[r]; pt += Dt[r]; pd += Dd[r]; }

    const float iou_sum = __shfl(ps, 0, 32) + __shfl(ps, 16, 32);
    const float tp_tot  = __shfl(pt, 0, 32) + __shfl(pt, 16, 32);  /* exact <2^24 */
    const float dn_tot  = __shfl(pd, 0, 32) + __shfl(pd, 16, 32);  /* exact <2^24 */

    if (l == 0) {
        const double total  = (double)BATCH * CLS * PLANE;
        const double tn_tot = total - (double)dn_tot;
        res[0] = iou_sum / (float)CLS;                        /* macro IoU  */
        res[1] = (float)(((double)tp_tot + tn_tot) / total);  /* pixel acc  */
    }
}

extern "C" void kernel_launch(void* const* d_in, const int* in_sizes, int n_in,
                              void* d_out, int out_size, void* d_ws, size_t ws_size,
                              hipStream_t stream) {
    (void)in_sizes; (void)n_in; (void)out_size; (void)ws_size;
    const float* outp = (const float*)d_in[0];
    const float* tgtp = (const float*)d_in[1];
    unsigned int* ws  = (unsigned int*)d_ws;
    float* res        = (float*)d_out;

    iou_zero_ws      <<<1, 64, 0, stream>>>(ws);
    iou_conf_hist_tdm<<<NBLOCK, NTHREAD, 0, stream>>>(outp, tgtp, ws);
    iou_finalize_wmma<<<1, 32, 0, stream>>>(ws, res);
}